// WeightedBCELoss_29961691857675
// MI455X (gfx1250) — compile-verified
//
#include <hip/hip_runtime.h>
#include <math.h>

// ---------------------------------------------------------------------------
// Fused boundary-weighted BCE for [16,1,1024,1024] f32.
//   weights = 1 + 2 * (dilate9x9(y) - erode9x9(y))      (BOUNDARY_WEIGHT = 3)
//   loss    = mean(weights * (max(x,0) - x*y + log1p(exp(-|x|))))
// HBM-bound (128 MB one-pass => ~5.5us @ 23.3 TB/s). VALU minimized via:
//   - hardware v_exp/v_log softplus
//   - 3x3-decomposed separable 9-tap min/max with shared 3-tap partials
//   - b128 global/LDS traffic (each thread owns a 16B-aligned 4-col strip)
// Wave reductions use V_WMMA_F32_16X16X4_F32 (exact f32, ones-B trick).
// ---------------------------------------------------------------------------

typedef __attribute__((ext_vector_type(2))) float v2f;
typedef __attribute__((ext_vector_type(4))) float v4f;
typedef __attribute__((ext_vector_type(8))) float v8f;

#define TILE    64
#define HALO    4
#define PADT    (TILE + 2 * HALO)   // 72
#define TSTRIDE 76                  // stride for s_tgt: mult of 4 (16B vec align)
#define HDIM    1024
#define WDIM    1024
#define NBATCH  16
#define NTOTAL  (NBATCH * HDIM * WDIM)  // 16,777,216

__device__ __forceinline__ v4f vmin4(v4f a, v4f b) {
    v4f r;
    #pragma unroll
    for (int i = 0; i < 4; ++i) r[i] = fminf(a[i], b[i]);
    return r;
}
__device__ __forceinline__ v4f vmax4(v4f a, v4f b) {
    v4f r;
    #pragma unroll
    for (int i = 0; i < 4; ++i) r[i] = fmaxf(a[i], b[i]);
    return r;
}

// Wave-level sum of (acc0 + acc1) across all 32 lanes using one f32 WMMA.
// A (16x4) = the wave's 64 accumulator values, B (4x16) = ones, C = 0.
// D[m][n] = rowsum_m (replicated over n). Per-lane sum of the 8 D VGPRs gives
// sum(rows 0..7) on lanes 0..15 and sum(rows 8..15) on lanes 16..31.
__device__ __forceinline__ float wave_reduce_wmma(float acc0, float acc1) {
    v2f a; a.x = acc0; a.y = acc1;
    v2f b; b.x = 1.0f; b.y = 1.0f;
    v8f c = {};
    c = __builtin_amdgcn_wmma_f32_16x16x4_f32(
        /*neg_a=*/false, a, /*neg_b=*/false, b,
        /*c_mod=*/(short)0, c, /*reuse_a=*/false, /*reuse_b=*/false);
    float s = c[0] + c[1] + c[2] + c[3] + c[4] + c[5] + c[6] + c[7];
    float lo = __shfl(s, 0, 32);
    float hi = __shfl(s, 16, 32);
    return lo + hi;
}

__global__ __launch_bounds__(256) void fused_boundary_bce_kernel(
    const float* __restrict__ logits,
    const float* __restrict__ targets,
    float* __restrict__ partials)
{
    __shared__ __align__(16) float s_tgt[PADT * TSTRIDE];   // 72 x 76 target tile
    __shared__ __align__(16) float s_hmin[PADT * TILE];     // 72 x 64 horiz 9-min
    __shared__ __align__(16) float s_hmax[PADT * TILE];     // 72 x 64 horiz 9-max
    __shared__ float s_wave[8];

    const int tid  = threadIdx.x;
    const int bx   = blockIdx.x;   // tile col
    const int by   = blockIdx.y;   // tile row
    const int bz   = blockIdx.z;   // batch
    const int col0 = bx * TILE - HALO;
    const int row0 = by * TILE - HALO;
    const float* img = targets + (size_t)bz * (HDIM * WDIM);
    const float* lg  = logits  + (size_t)bz * (HDIM * WDIM);

    // ---- Load 72x72 target tile into LDS. --------------------------------
    // Interior blocks: b128 fast path (col0*4 = 256*bx-16 is 16B aligned).
    // Edge blocks: scalar clamp-to-edge (== the reference's valid-only op).
    const bool interior = (row0 >= 0) && (col0 >= 0) &&
                          (row0 + PADT <= HDIM) && (col0 + PADT <= WDIM);
    if (interior) {
        for (int i = tid; i < PADT * 18; i += 256) {   // 72 rows x 18 vec4
            int r = i / 18, c4 = (i - r * 18) << 2;
            v4f v = *(const v4f*)&img[(size_t)(row0 + r) * WDIM + (col0 + c4)];
            *(v4f*)&s_tgt[r * TSTRIDE + c4] = v;
        }
    } else {
        for (int i = tid; i < PADT * PADT; i += 256) {
            int r = i / PADT, c = i - r * PADT;
            int gr = row0 + r; gr = gr < 0 ? 0 : (gr > HDIM - 1 ? HDIM - 1 : gr);
            int gc = col0 + c; gc = gc < 0 ? 0 : (gc > WDIM - 1 ? WDIM - 1 : gc);
            s_tgt[r * TSTRIDE + c] = img[(size_t)gr * WDIM + gc];
        }
    }
    __syncthreads();

    // ---- Horizontal 9-tap min/max, 3x3 decomposed. -----------------------
    // Item = (row r, vec4 out-col vc): needs input cols 4vc..4vc+11 (3xb128).
    // out[j] (j=0..3) = op(m3[j+1], m3[j+4], m3[j+7]), m3 shared across j.
    for (int i = tid; i < PADT * 16; i += 256) {
        int r = i >> 4, vc4 = (i & 15) << 2;
        const float* base = &s_tgt[r * TSTRIDE + vc4];
        v4f e0 = *(const v4f*)(base + 0);
        v4f e1 = *(const v4f*)(base + 4);
        v4f e2 = *(const v4f*)(base + 8);
        float e[12];
        #pragma unroll
        for (int k = 0; k < 4; ++k) { e[k] = e0[k]; e[k+4] = e1[k]; e[k+8] = e2[k]; }
        float m3n[11], m3x[11];                 // centers 1..10
        #pragma unroll
        for (int p = 1; p <= 10; ++p) {
            m3n[p] = fminf(fminf(e[p-1], e[p]), e[p+1]);
            m3x[p] = fmaxf(fmaxf(e[p-1], e[p]), e[p+1]);
        }
        v4f on, ox;
        #pragma unroll
        for (int j = 0; j < 4; ++j) {
            on[j] = fminf(fminf(m3n[j+1], m3n[j+4]), m3n[j+7]);
            ox[j] = fmaxf(fmaxf(m3x[j+1], m3x[j+4]), m3x[j+7]);
        }
        *(v4f*)&s_hmin[r * TILE + vc4] = on;
        *(v4f*)&s_hmax[r * TILE + vc4] = ox;
    }
    __syncthreads();

    // ---- Vertical 9-tap (3x3 decomposed) + fused weighted BCE. -----------
    // Each thread owns a 4-row x 4-col block: 256 threads cover the full
    // 64x64 tile in one shot. 12 b128 row-loads feed 16 outputs per array.
    const int vc4 = (tid & 15) << 2;          // cols vc4..vc4+3
    const int r0  = (tid >> 4) << 2;          // rows r0..r0+3
    float acc0 = 0.0f, acc1 = 0.0f;           // two K-slots of the WMMA A matrix

    v4f cmn[12], cmx[12];
    #pragma unroll
    for (int k = 0; k < 12; ++k) {
        cmn[k] = *(const v4f*)&s_hmin[(r0 + k) * TILE + vc4];
        cmx[k] = *(const v4f*)&s_hmax[(r0 + k) * TILE + vc4];
    }
    v4f r3n[11], r3x[11];                     // row-3-tap centers 1..10
    #pragma unroll
    for (int q = 1; q <= 10; ++q) {
        r3n[q] = vmin4(vmin4(cmn[q-1], cmn[q]), cmn[q+1]);
        r3x[q] = vmax4(vmax4(cmx[q-1], cmx[q]), cmx[q+1]);
    }

    const int grow0 = by * TILE + r0;
    const int gcol  = bx * TILE + vc4;
    #pragma unroll
    for (int j = 0; j < 4; ++j) {
        v4f vmn = vmin4(vmin4(r3n[j+1], r3n[j+4]), r3n[j+7]);
        v4f vmx = vmax4(vmax4(r3x[j+1], r3x[j+4]), r3x[j+7]);
        size_t gidx = (size_t)(grow0 + j) * WDIM + gcol;
        v4f x4 = *(const v4f*)&lg[gidx];
        v4f y4 = *(const v4f*)&img[gidx];
        float vsum = 0.0f;
        #pragma unroll
        for (int k = 0; k < 4; ++k) {
            float wgt = 1.0f + 2.0f * (vmx[k] - vmn[k]);   // 1 + (BW-1)*boundary
            float x = x4[k], y = y4[k];
            // stable BCE-with-logits; softplus via hardware v_exp/v_log
            float per = fmaxf(x, 0.0f) - x * y + __logf(1.0f + __expf(-fabsf(x)));
            vsum += wgt * per;
        }
        if (j & 1) acc1 += vsum; else acc0 += vsum;
    }

    // ---- Block reduction: WMMA per wave, then 8 wave totals in LDS. ------
    float wsum = wave_reduce_wmma(acc0, acc1);
    int lane = tid & 31;
    int wid  = tid >> 5;
    if (lane == 0) s_wave[wid] = wsum;
    __syncthreads();
    if (tid == 0) {
        float t = 0.0f;
        #pragma unroll
        for (int i = 0; i < 8; ++i) t += s_wave[i];
        partials[((size_t)bz * gridDim.y + by) * gridDim.x + bx] = t;
    }
}

// Deterministic final reduction of the 4096 block partials -> mean.
__global__ __launch_bounds__(256) void reduce_partials_kernel(
    const float* __restrict__ partials, float* __restrict__ out, int n)
{
    __shared__ float s_wave[8];
    float acc0 = 0.0f, acc1 = 0.0f;
    int i = threadIdx.x;
    for (; i + 256 < n; i += 512) {
        acc0 += partials[i];
        acc1 += partials[i + 256];
    }
    if (i < n) acc0 += partials[i];

    float wsum = wave_reduce_wmma(acc0, acc1);
    int lane = threadIdx.x & 31;
    int wid  = threadIdx.x >> 5;
    if (lane == 0) s_wave[wid] = wsum;
    __syncthreads();
    if (threadIdx.x == 0) {
        float t = 0.0f;
        #pragma unroll
        for (int k = 0; k < 8; ++k) t += s_wave[k];
        out[0] = t * (1.0f / (float)NTOTAL);
    }
}

extern "C" void kernel_launch(void* const* d_in, const int* in_sizes, int n_in,
                              void* d_out, int out_size, void* d_ws, size_t ws_size,
                              hipStream_t stream) {
    const float* logits  = (const float*)d_in[0];
    const float* targets = (const float*)d_in[1];
    float* partials = (float*)d_ws;           // 16*16*16 = 4096 floats (16 KB)

    dim3 grid(WDIM / TILE, HDIM / TILE, NBATCH);   // 16 x 16 x 16 = 4096 blocks
    fused_boundary_bce_kernel<<<grid, 256, 0, stream>>>(logits, targets, partials);

    int nblocks = (WDIM / TILE) * (HDIM / TILE) * NBATCH;
    reduce_partials_kernel<<<1, 256, 0, stream>>>(partials, (float*)d_out, nblocks);
}